// VectorQuantizer_75084618268725
// MI455X (gfx1250) — compile-verified
//
#include <hip/hip_runtime.h>
#include <hip/hip_bf16.h>
#include <stdint.h>

// CDNA5 (gfx1250) VQ-VAE vector quantizer.
//   inputs:  d_in[0] = X [16,4096,256] f32  (N=65536 rows, D=256)
//            d_in[1] = E [1024,256]    f32  (K=1024 codes)
//   output:  d_out = [loss(1 f32), quantized(16777216 f32)]
// Distance GEMM (N x K, reduce D) on V_WMMA_F32_16X16X4_F32 (exact fp32).
// 32-code tiles double-buffered in LDS via GLOBAL_LOAD_ASYNC_TO_LDS_B128
// (ASYNCcnt pipeline: next tile streams in while current tile feeds WMMAs).
// Argmin is branchless (v_cndmask) so EXEC stays all-ones around WMMA;
// final 16-lane shfl_xor butterfly does min-with-first-index (jnp.argmin).

typedef float v2f __attribute__((ext_vector_type(2)));
typedef float v8f __attribute__((ext_vector_type(8)));

#define VQ_N       65536
#define VQ_D       256
#define VQ_K       1024
#define LDS_STRIDE 260   // 256 + 4 dword pad: 16B-aligned rows, distinct banks
#define NTILES     (VQ_K / 32)

// ---------------------------------------------------------------- ||e_k||^2
__global__ __launch_bounds__(256) void vq_enorm_kernel(
    const float* __restrict__ E, float* __restrict__ enorm) {
  int k = blockIdx.x * 256 + threadIdx.x;
  if (k < VQ_K) {
    const float4* row = (const float4*)(E + (size_t)k * VQ_D);
    float s = 0.f;
#pragma unroll 4
    for (int i = 0; i < VQ_D / 4; ++i) {
      float4 v = row[i];
      s = fmaf(v.x, v.x, fmaf(v.y, v.y, fmaf(v.z, v.z, fmaf(v.w, v.w, s))));
    }
    enorm[k] = s;
  }
}

// Issue one 32x256 f32 code tile as 8 async B128 loads per thread.
// gbase: this thread's first source row slice; ldsbyte: matching LDS dest.
__device__ __forceinline__ void vq_stage_tile_async(const float* gbase,
                                                    uint32_t ldsbyte) {
#pragma unroll
  for (int i = 0; i < 8; ++i) {
    const float* g = gbase + (size_t)i * 4 * VQ_D;      // 4 code rows / step
    uint32_t l = ldsbyte + (uint32_t)(i * 4 * LDS_STRIDE * 4);
    asm volatile("global_load_async_to_lds_b128 %0, %1, off"
                 :: "v"(l), "v"(g) : "memory");
  }
}

// -------------------------------------------------- argmin over codebook
// Block = 256 threads = 8 waves; wave w owns rows [blk*128 + w*16, +16).
// 32 iterations over 32-code tiles; per tile: 2 x 64 fp32 WMMAs.
__global__ __launch_bounds__(256) void vq_argmin_kernel(
    const float* __restrict__ X, const float* __restrict__ E,
    const float* __restrict__ enorm, int* __restrict__ indices) {
  __shared__ float sE[2][32 * LDS_STRIDE];  // 2 x 33.3 KB double buffer

  const int tid   = threadIdx.x;
  const int lane  = tid & 31;
  const int wave  = tid >> 5;
  const int lrow  = lane & 15;       // A: matrix row   B: matrix col (code)
  const int khalf = lane >> 4;       // which K-pair of the 16x4 fragment
  const int rowbase = blockIdx.x * 128 + wave * 16;

  // A fragment source: lane reads X[rowbase+lrow][4*s + 2*khalf .. +1]
  const float* aRow = X + (size_t)(rowbase + lrow) * VQ_D + khalf * 2;

  // Per-thread staging coordinates: row r0 = tid>>6 (+4 per step),
  // 16-byte column group cg = tid&63.
  const int r0 = tid >> 6;
  const int cg = tid & 63;
  const float* gstage = E + (size_t)r0 * VQ_D + cg * 4;
  const uint32_t lds0 = (uint32_t)(size_t)&sE[0][0] +
                        (uint32_t)(r0 * LDS_STRIDE * 4 + cg * 16);
  const uint32_t BUFBYTES = 32 * LDS_STRIDE * 4;

  float bd[8];
  int   bi[8];
#pragma unroll
  for (int j = 0; j < 8; ++j) { bd[j] = 3.4e38f; bi[j] = 0; }

  // Prime the pipeline: tile 0 -> buffer 0.
  vq_stage_tile_async(gstage, lds0);

  for (int ct = 0; ct < NTILES; ++ct) {
    const int cur = ct & 1;
    if (ct + 1 < NTILES) {
      // Stream next tile into the other buffer while we compute.
      vq_stage_tile_async(gstage + (size_t)(ct + 1) * 32 * VQ_D,
                          lds0 + (cur ^ 1) * BUFBYTES);
      // Async loads complete in order: <=8 outstanding means tile ct landed.
      asm volatile("s_wait_asynccnt 0x8" ::: "memory");
    } else {
      asm volatile("s_wait_asynccnt 0x0" ::: "memory");
    }
    __syncthreads();  // tile ct visible to all waves

    // G = Xtile(16xD) * Etile^T(Dx32): two 16x16 outputs share each A frag.
    v8f acc0 = {};
    v8f acc1 = {};
    const float* bRow0 = &sE[cur][lrow * LDS_STRIDE + khalf * 2];
    const float* bRow1 = &sE[cur][(lrow + 16) * LDS_STRIDE + khalf * 2];
#pragma unroll 8
    for (int s = 0; s < VQ_D / 4; ++s) {
      v2f a  = *(const v2f*)(aRow + s * 4);
      v2f b0 = *(const v2f*)(bRow0 + s * 4);
      v2f b1 = *(const v2f*)(bRow1 + s * 4);
      acc0 = __builtin_amdgcn_wmma_f32_16x16x4_f32(
          false, a, false, b0, (short)0, acc0, false, false);
      acc1 = __builtin_amdgcn_wmma_f32_16x16x4_f32(
          false, a, false, b1, (short)0, acc1, false, false);
    }

    // dist(row, k) = ||e_k||^2 - 2*G  (||x||^2 constant per row -> dropped).
    // C/D layout: slot j <-> row khalf*8 + j, column fixed = lrow.
    // Branchless: k grows monotonically per lane, so strict '<' keeps the
    // first (lowest-index) minimum. Update k0 tile before k1.
    const int   k0  = ct * 32 + lrow;
    const int   k1  = k0 + 16;
    const float en0 = enorm[k0];
    const float en1 = enorm[k1];
#pragma unroll
    for (int j = 0; j < 8; ++j) {
      float d0 = fmaf(-2.0f, acc0[j], en0);
      bool lt0 = d0 < bd[j];
      bi[j] = lt0 ? k0 : bi[j];
      bd[j] = lt0 ? d0 : bd[j];
      float d1 = fmaf(-2.0f, acc1[j], en1);
      bool lt1 = d1 < bd[j];
      bi[j] = lt1 ? k1 : bi[j];
      bd[j] = lt1 ? d1 : bd[j];
    }
    __syncthreads();  // everyone done with buf cur before it is re-staged
  }

  // Reduce (min, first-index) across the 16 lanes holding one row.
  // xor of bits 0..3 stays inside each 16-lane half of the wave32.
#pragma unroll
  for (int j = 0; j < 8; ++j) {
#pragma unroll
    for (int off = 8; off >= 1; off >>= 1) {
      float od = __shfl_xor(bd[j], off, 32);
      int   oi = __shfl_xor(bi[j], off, 32);
      bool take = (od < bd[j]) || (od == bd[j] && oi < bi[j]);
      bd[j] = take ? od : bd[j];
      bi[j] = take ? oi : bi[j];
    }
  }
  if (lrow == 0) {
#pragma unroll
    for (int j = 0; j < 8; ++j)
      indices[rowbase + khalf * 8 + j] = bi[j];
  }
}

// ---------------------------------- gather codebook rows + loss partials
// 1024 blocks x 256 threads; block b handles rows [b*64, b*64+64),
// thread t handles dim t. Deterministic per-block tree reduction.
__global__ __launch_bounds__(256) void vq_gather_kernel(
    const float* __restrict__ X, const float* __restrict__ E,
    const int* __restrict__ indices, float* __restrict__ quant,
    float* __restrict__ partial) {
  __shared__ float red[256];
  const int t = threadIdx.x;
  const int b = blockIdx.x;
  float acc = 0.f;
#pragma unroll 4
  for (int r = 0; r < 64; ++r) {
    int row = b * 64 + r;
    int idx = indices[row];
    float e = E[(size_t)idx * VQ_D + t];
    float x = X[(size_t)row * VQ_D + t];
    float dlt = e - x;
    acc = fmaf(dlt, dlt, acc);
    quant[(size_t)row * VQ_D + t] = e;   // straight-through output == e
  }
  red[t] = acc;
  __syncthreads();
  for (int s = 128; s > 0; s >>= 1) {
    if (t < s) red[t] += red[t + s];
    __syncthreads();
  }
  if (t == 0) partial[b] = red[0];
}

// ------------------------------------------------------------- finalize
__global__ __launch_bounds__(256) void vq_finalize_kernel(
    const float* __restrict__ partial, float* __restrict__ loss) {
  __shared__ float red[256];
  const int t = threadIdx.x;
  red[t] = partial[t] + partial[t + 256] + partial[t + 512] + partial[t + 768];
  __syncthreads();
  for (int s = 128; s > 0; s >>= 1) {
    if (t < s) red[t] += red[t + s];
    __syncthreads();
  }
  // q_latent_loss + 0.25*e_latent_loss == 1.25 * MSE (numerically)
  if (t == 0) loss[0] = 1.25f * red[0] / 16777216.0f;
}

extern "C" void kernel_launch(void* const* d_in, const int* in_sizes, int n_in,
                              void* d_out, int out_size, void* d_ws, size_t ws_size,
                              hipStream_t stream) {
  const float* X = (const float*)d_in[0];   // [65536, 256]
  const float* E = (const float*)d_in[1];   // [1024, 256]
  float* out   = (float*)d_out;             // [1 + 16777216]
  float* loss  = out;
  float* quant = out + 1;

  // workspace layout (floats): enorm[1024] | indices[65536] (int) | partial[1024]
  float* enorm   = (float*)d_ws;
  int*   indices = (int*)d_ws + VQ_K;
  float* partial = (float*)d_ws + VQ_K + VQ_N;

  vq_enorm_kernel<<<VQ_K / 256, 256, 0, stream>>>(E, enorm);
  vq_argmin_kernel<<<VQ_N / 128, 256, 0, stream>>>(X, E, enorm, indices);
  vq_gather_kernel<<<VQ_N / 64, 256, 0, stream>>>(X, E, indices, quant, partial);
  vq_finalize_kernel<<<1, 256, 0, stream>>>(partial, loss);
}